// naludnc_5488968204595
// MI455X (gfx1250) — compile-verified
//
#include <hip/hip_runtime.h>
#include <hip/hip_bf16.h>
#include <math.h>

typedef float v2f __attribute__((ext_vector_type(2)));
typedef float v8f __attribute__((ext_vector_type(8)));

#define N_MEM 512
#define WD    64
#define RH    4
#define OUTD  256
#define IFACE 471
#define CTRL  727
#define BS    64

// ---------------- workspace layout (float offsets) ----------------
constexpr size_t WS_IFW    = 0;
constexpr size_t WS_IFR    = WS_IFW    + (size_t)BS * IFACE;
constexpr size_t WS_KWN    = WS_IFR    + (size_t)BS * IFACE;
constexpr size_t WS_BW     = WS_KWN    + (size_t)BS * WD;
constexpr size_t WS_ERASE  = WS_BW     + (size_t)BS;
constexpr size_t WS_WV     = WS_ERASE  + (size_t)BS * WD;
constexpr size_t WS_FG     = WS_WV     + (size_t)BS * WD;
constexpr size_t WS_AG     = WS_FG     + (size_t)BS * RH;
constexpr size_t WS_WG     = WS_AG     + (size_t)BS;
constexpr size_t WS_KRN    = WS_WG     + (size_t)BS;
constexpr size_t WS_BR     = WS_KRN    + (size_t)BS * RH * WD;
constexpr size_t WS_RM     = WS_BR     + (size_t)BS * RH;
constexpr size_t WS_WALLOC = WS_RM     + (size_t)BS * 12;
constexpr size_t WS_WFWD   = WS_WALLOC + (size_t)BS * N_MEM;
constexpr size_t WS_WBWD   = WS_WFWD   + (size_t)BS * N_MEM * RH;
constexpr size_t WS_READV  = WS_WBWD   + (size_t)BS * N_MEM * RH;

// ---------------- output layout (float offsets, tuple order) ----------------
constexpr size_t O_Y   = 0;
constexpr size_t O_M   = O_Y   + (size_t)BS * OUTD;
constexpr size_t O_U   = O_M   + (size_t)BS * N_MEM * WD;
constexpr size_t O_L   = O_U   + (size_t)BS * N_MEM;
constexpr size_t O_WP  = O_L   + (size_t)BS * N_MEM * N_MEM;
constexpr size_t O_WRN = O_WP  + (size_t)BS * N_MEM;
constexpr size_t O_WW  = O_WRN + (size_t)BS * N_MEM * RH;

__device__ __forceinline__ float sigmoidf_(float x) { return 1.f / (1.f + expf(-x)); }
__device__ __forceinline__ float softplusf_(float x) {
    return fmaxf(x, 0.f) + log1pf(expf(-fabsf(x)));
}

// ---------------- generic fp32 WMMA GEMM: C[M,N] = A[M,K] * B[K,N] ----------------
// One wave per 16x16 tile, V_WMMA_F32_16X16X4_F32 over K.
// Rows (M) must be a multiple of 16 (true for all call sites).
// Ragged N handled by clamping the B column pointer: B column n feeds only
// D column n, and OOB columns are never stored, so no masking is needed.
// Ragged K handled by a single masked tail step (branch-free selects).
__global__ void gemm_wmma_f32(const float* __restrict__ A, const float* __restrict__ B,
                              float* __restrict__ C, int Ncols, int K,
                              int lda, int ldb, int ldc)
{
    const int n0   = blockIdx.x * 16;
    const int m0   = blockIdx.y * 16;
    const int lane = threadIdx.x;        // 0..31, wave32
    const int lo   = lane & 15;
    const int hi   = lane >> 4;
    const int col  = n0 + lo;
    const bool colok = (col < Ncols);

    const float* Arow = A + (size_t)(m0 + lo) * lda + 2 * hi;  // lane's k-pair base
    const float* Bcol = B + (colok ? col : 0) + (size_t)(2 * hi) * ldb;

    v8f acc = {0.f, 0.f, 0.f, 0.f, 0.f, 0.f, 0.f, 0.f};
    const int K4 = K & ~3;
    // main loop: no bounds checks at all
    for (int k0 = 0; k0 < K4; k0 += 4) {
        v2f a, b;
        a.x = Arow[k0];
        a.y = Arow[k0 + 1];
        b.x = Bcol[(size_t)k0 * ldb];
        b.y = Bcol[(size_t)(k0 + 1) * ldb];
        acc = __builtin_amdgcn_wmma_f32_16x16x4_f32(false, a, false, b,
                                                    (short)0, acc, false, false);
    }
    // masked tail (K % 4 != 0): clamp addresses, zero via selects (no branches)
    if (K4 < K) {
        const int ka  = K4 + 2 * hi;
        const int kc0 = (ka     < K) ? ka     : 0;
        const int kc1 = (ka + 1 < K) ? ka + 1 : 0;
        const float m0f = (ka     < K) ? 1.f : 0.f;
        const float m1f = (ka + 1 < K) ? 1.f : 0.f;
        const float* Ar0 = Arow - 2 * hi;     // back to row base
        const float* Bc0 = Bcol - (size_t)(2 * hi) * ldb;
        v2f a, b;
        a.x = Ar0[kc0] * m0f;                 // mask A side only; zero kills product
        a.y = Ar0[kc1] * m1f;
        b.x = Bc0[(size_t)kc0 * ldb];
        b.y = Bc0[(size_t)kc1 * ldb];
        acc = __builtin_amdgcn_wmma_f32_16x16x4_f32(false, a, false, b,
                                                    (short)0, acc, false, false);
    }
    if (colok) {
#pragma unroll
        for (int v = 0; v < 8; ++v)       // VGPR v: M=v (lanes 0-15), M=v+8 (lanes 16-31)
            C[(size_t)(m0 + v + 8 * hi) * ldc + col] = acc[v];
    }
}

// ---------------- interface partition + gates + key normalization ----------------
__global__ void prep_kernel(float* __restrict__ ws)
{
    __shared__ float red[64];
    const int b = blockIdx.x;
    const int t = threadIdx.x;           // 64 threads
    const float* ifw = ws + WS_IFW + (size_t)b * IFACE;
    const float* ifr = ws + WS_IFR + (size_t)b * IFACE;

    // k_write: iface[260..323], l2-normalize
    float kwv = ifw[260 + t];
    red[t] = kwv * kwv; __syncthreads();
    for (int s = 32; s > 0; s >>= 1) { if (t < s) red[t] += red[t + s]; __syncthreads(); }
    ws[WS_KWN + b * WD + t] = kwv * rsqrtf(fmaxf(red[0], 1e-12f));

    ws[WS_ERASE + b * WD + t] = sigmoidf_(ifw[325 + t]);
    ws[WS_WV    + b * WD + t] = ifw[389 + t];
    if (t == 0) {
        ws[WS_BW + b] = 1.f + softplusf_(ifw[324]);
        ws[WS_AG + b] = sigmoidf_(ifw[457]);
        ws[WS_WG + b] = sigmoidf_(ifw[458]);
    }
    if (t < RH) ws[WS_FG + b * RH + t] = sigmoidf_(ifw[453 + t]);

    // k_read heads: iface[r*64 .. r*64+63], l2-normalize each
    for (int r = 0; r < RH; ++r) {
        float kv = ifr[r * WD + t];
        __syncthreads();
        red[t] = kv * kv; __syncthreads();
        for (int s = 32; s > 0; s >>= 1) { if (t < s) red[t] += red[t + s]; __syncthreads(); }
        ws[WS_KRN + b * RH * WD + r * WD + t] = kv * rsqrtf(fmaxf(red[0], 1e-12f));
    }
    if (t < RH) {
        ws[WS_BR + b * RH + t] = 1.f + softplusf_(ifr[256 + t]);
        // read_modes: softmax over the 3 modes, column r=t
        float m0 = ifr[459 + t], m1 = ifr[459 + RH + t], m2 = ifr[459 + 2 * RH + t];
        float mx = fmaxf(m0, fmaxf(m1, m2));
        float e0 = expf(m0 - mx), e1 = expf(m1 - mx), e2 = expf(m2 - mx);
        float s = e0 + e1 + e2;
        ws[WS_RM + b * 12 + t]          = e0 / s;
        ws[WS_RM + b * 12 + RH + t]     = e1 / s;
        ws[WS_RM + b * 12 + 2 * RH + t] = e2 / s;
    }
}

// ---------------- usage update (retention) ----------------
__global__ void usage_kernel(const float* __restrict__ usage_in, const float* __restrict__ Wread,
                             const float* __restrict__ Wwrite, const float* __restrict__ fg,
                             float* __restrict__ usage_out)
{
    int idx = blockIdx.x * blockDim.x + threadIdx.x;
    if (idx >= BS * N_MEM) return;
    int b = idx >> 9;
    float ret = 1.f;
#pragma unroll
    for (int r = 0; r < RH; ++r)
        ret *= 1.f - fg[b * RH + r] * Wread[(size_t)idx * RH + r];
    float u = usage_in[idx], w = Wwrite[idx];
    usage_out[idx] = (u + w - u * w) * ret;
}

// ---------------- allocation weighting: bitonic sort + product scan + scatter ----------------
__global__ void alloc_kernel(const float* __restrict__ usage, float* __restrict__ Walloc)
{
    __shared__ float su[N_MEM];
    __shared__ int   si[N_MEM];
    __shared__ float cp[N_MEM];
    const int b = blockIdx.x;
    const int t = threadIdx.x;           // 512 threads
    su[t] = usage[b * N_MEM + t];
    si[t] = t;
    __syncthreads();
    // ascending by (usage, index) -> matches top_k(-u) stability
    for (int k = 2; k <= N_MEM; k <<= 1) {
        for (int j = k >> 1; j > 0; j >>= 1) {
            int ixj = t ^ j;
            if (ixj > t) {
                float a = su[t], c = su[ixj];
                int ia = si[t], ic = si[ixj];
                bool agt = (a > c) || (a == c && ia > ic);
                bool up = ((t & k) == 0);
                if (up == agt) { su[t] = c; su[ixj] = a; si[t] = ic; si[ixj] = ia; }
            }
            __syncthreads();
        }
    }
    // inclusive product scan of sorted usage
    cp[t] = su[t];
    __syncthreads();
    for (int off = 1; off < N_MEM; off <<= 1) {
        float v = cp[t];
        float p = (t >= off) ? cp[t - off] : 1.f;
        __syncthreads();
        cp[t] = v * p;
        __syncthreads();
    }
    float excl = (t == 0) ? 1.f : cp[t - 1];
    Walloc[b * N_MEM + si[t]] = (1.f - su[t]) * excl;   // permutation scatter
}

// ---------------- write path: content lookup, Ww, precedence, M rewrite ----------------
__global__ void write_kernel(const float* __restrict__ M_in, const float* __restrict__ Wp_in,
                             float* __restrict__ ws, float* __restrict__ out)
{
    __shared__ float kw[WD], er[WD], wv[WD];
    __shared__ float sim[N_MEM];
    __shared__ float wwl[N_MEM];
    __shared__ float red[256];
    const int b = blockIdx.x;
    const int t = threadIdx.x;           // 256 threads
    if (t < WD) {
        kw[t] = ws[WS_KWN   + b * WD + t];
        er[t] = ws[WS_ERASE + b * WD + t];
        wv[t] = ws[WS_WV    + b * WD + t];
    }
    __syncthreads();
    const float bw = ws[WS_BW + b], ag = ws[WS_AG + b], wg = ws[WS_WG + b];
    const float* Mb = M_in + (size_t)b * N_MEM * WD;
#pragma unroll
    for (int h = 0; h < 2; ++h) {
        const int n = t + h * 256;
        const float* Mr = Mb + n * WD;
        float ss = 0.f, dot = 0.f;
        for (int w = 0; w < WD; ++w) { float m = Mr[w]; ss += m * m; dot += m * kw[w]; }
        sim[n] = dot * rsqrtf(fmaxf(ss, 1e-12f)) * bw;
    }
    __syncthreads();
    red[t] = fmaxf(sim[t], sim[t + 256]);
    __syncthreads();
    for (int s = 128; s > 0; s >>= 1) { if (t < s) red[t] = fmaxf(red[t], red[t + s]); __syncthreads(); }
    const float mx = red[0];
    __syncthreads();
    float e0 = expf(sim[t] - mx), e1 = expf(sim[t + 256] - mx);
    red[t] = e0 + e1;
    __syncthreads();
    for (int s = 128; s > 0; s >>= 1) { if (t < s) red[t] += red[t + s]; __syncthreads(); }
    const float inv_s = 1.f / red[0];
    __syncthreads();
    float wa0 = ws[WS_WALLOC + b * N_MEM + t], wa1 = ws[WS_WALLOC + b * N_MEM + t + 256];
    float ww0 = wg * (ag * wa0 + (1.f - ag) * e0 * inv_s);
    float ww1 = wg * (ag * wa1 + (1.f - ag) * e1 * inv_s);
    wwl[t] = ww0; wwl[t + 256] = ww1;
    out[O_WW + b * N_MEM + t] = ww0;
    out[O_WW + b * N_MEM + t + 256] = ww1;
    red[t] = ww0 + ww1;
    __syncthreads();
    for (int s = 128; s > 0; s >>= 1) { if (t < s) red[t] += red[t + s]; __syncthreads(); }
    const float sww = red[0];
    {   // precedence update
        float wp0 = Wp_in[b * N_MEM + t], wp1 = Wp_in[b * N_MEM + t + 256];
        out[O_WP + b * N_MEM + t]       = (1.f - sww) * wp0 + ww0;
        out[O_WP + b * N_MEM + t + 256] = (1.f - sww) * wp1 + ww1;
    }
    // M rewrite
    for (int idx = t; idx < N_MEM * WD; idx += 256) {
        int n = idx >> 6, w = idx & 63;
        float m = Mb[idx];
        out[O_M + (size_t)b * N_MEM * WD + idx] =
            m * (1.f - wwl[n] * er[w]) + wwl[n] * wv[w];
    }
}

// ---------------- zero helper ----------------
__global__ void zero_kernel(float* __restrict__ p, int n)
{
    int i = blockIdx.x * blockDim.x + threadIdx.x;
    if (i < n) p[i] = 0.f;
}

// ---------------- L update fused with L@Wr and L^T@Wr (single pass over L) ----------------
__global__ void l_kernel(const float* __restrict__ L_in, const float* __restrict__ Wp_in,
                         const float* __restrict__ Wr_in, const float* __restrict__ Ww,
                         float* __restrict__ L_out, float* __restrict__ Wfwd,
                         float* __restrict__ Wbwd)
{
    __shared__ float wwi[64];
    __shared__ float wri[64 * RH];
    __shared__ float fwd[64 * RH];
    const int b  = blockIdx.y;
    const int i0 = blockIdx.x * 64;
    const int t  = threadIdx.x;          // 256 threads
    if (t < 64) wwi[t] = Ww[b * N_MEM + i0 + t];
    wri[t] = Wr_in[(size_t)b * N_MEM * RH + i0 * RH + t];
    fwd[t] = 0.f;
    __syncthreads();
    const int j1 = t, j2 = t + 256;      // columns owned by this thread
    const float wwj1 = Ww[b * N_MEM + j1],    wwj2 = Ww[b * N_MEM + j2];
    const float wpj1 = Wp_in[b * N_MEM + j1], wpj2 = Wp_in[b * N_MEM + j2];
    float wrj1[RH], wrj2[RH];
#pragma unroll
    for (int r = 0; r < RH; ++r) {
        wrj1[r] = Wr_in[(size_t)b * N_MEM * RH + j1 * RH + r];
        wrj2[r] = Wr_in[(size_t)b * N_MEM * RH + j2 * RH + r];
    }
    float bwd1[RH] = {0.f, 0.f, 0.f, 0.f};
    float bwd2[RH] = {0.f, 0.f, 0.f, 0.f};
    const float* Lb = L_in  + (size_t)b * N_MEM * N_MEM;
    float*       Lo = L_out + (size_t)b * N_MEM * N_MEM;
    for (int ii = 0; ii < 64; ++ii) {
        const int i = i0 + ii;
        const float wi = wwi[ii];
        const float* Lr = Lb + (size_t)i * N_MEM;
        if (ii + 1 < 64) {                       // prefetch next L row
            __builtin_prefetch(Lr + N_MEM + j1, 0, 1);
            __builtin_prefetch(Lr + N_MEM + j2, 0, 1);
        }
        float l1 = Lr[j1], l2 = Lr[j2];
        float n1 = (1.f - wi - wwj1) * l1 + wi * wpj1;
        float n2 = (1.f - wi - wwj2) * l2 + wi * wpj2;
        if (i == j1) n1 = 0.f;
        if (i == j2) n2 = 0.f;
        Lo[(size_t)i * N_MEM + j1] = n1;
        Lo[(size_t)i * N_MEM + j2] = n2;
#pragma unroll
        for (int r = 0; r < RH; ++r) {
            float f = n1 * wrj1[r] + n2 * wrj2[r];       // W_fwd row contribution
            for (int m = 16; m > 0; m >>= 1) f += __shfl_xor(f, m, 32);
            if ((t & 31) == 0) atomicAdd(&fwd[ii * RH + r], f);   // ds_add_f32
            const float wir = wri[ii * RH + r];
            bwd1[r] += n1 * wir;                          // W_bwd column accumulators
            bwd2[r] += n2 * wir;
        }
    }
    __syncthreads();
    Wfwd[(size_t)b * N_MEM * RH + i0 * RH + t] = fwd[t];
#pragma unroll
    for (int r = 0; r < RH; ++r) {
        atomicAdd(&Wbwd[(size_t)b * N_MEM * RH + j1 * RH + r], bwd1[r]);
        atomicAdd(&Wbwd[(size_t)b * N_MEM * RH + j2 * RH + r], bwd2[r]);
    }
}

// ---------------- read path: lookup on new M, 4 softmaxes, W_read_n, read_v ----------------
__global__ void read_kernel(const float* __restrict__ M_new, float* __restrict__ ws,
                            float* __restrict__ out)
{
    __shared__ float kr[RH * WD];
    __shared__ float br[RH];
    __shared__ float rm[3 * RH];
    __shared__ float sim[N_MEM * RH];    // raw scores -> exp -> W_read_n
    __shared__ float red[256];
    __shared__ float ssum[RH];
    __shared__ float rv[4 * RH * WD];
    const int b = blockIdx.x;
    const int t = threadIdx.x;           // 256 threads
    kr[t] = ws[WS_KRN + (size_t)b * RH * WD + t];
    if (t < RH) br[t] = ws[WS_BR + b * RH + t];
    if (t < 3 * RH) rm[t] = ws[WS_RM + b * 12 + t];
    __syncthreads();
    const float* Mb = M_new + (size_t)b * N_MEM * WD;
#pragma unroll
    for (int h = 0; h < 2; ++h) {
        const int n = t + h * 256;
        const float* Mr = Mb + n * WD;
        float ss = 0.f, d0 = 0.f, d1 = 0.f, d2 = 0.f, d3 = 0.f;
        for (int w = 0; w < WD; ++w) {
            float m = Mr[w];
            ss += m * m;
            d0 += m * kr[0 * WD + w]; d1 += m * kr[1 * WD + w];
            d2 += m * kr[2 * WD + w]; d3 += m * kr[3 * WD + w];
        }
        float inv = rsqrtf(fmaxf(ss, 1e-12f));
        sim[n * RH + 0] = d0 * inv * br[0];
        sim[n * RH + 1] = d1 * inv * br[1];
        sim[n * RH + 2] = d2 * inv * br[2];
        sim[n * RH + 3] = d3 * inv * br[3];
    }
    __syncthreads();
    for (int r = 0; r < RH; ++r) {       // softmax over N per head
        red[t] = fmaxf(sim[t * RH + r], sim[(t + 256) * RH + r]);
        __syncthreads();
        for (int s = 128; s > 0; s >>= 1) { if (t < s) red[t] = fmaxf(red[t], red[t + s]); __syncthreads(); }
        const float mx = red[0];
        __syncthreads();
        float e0 = expf(sim[t * RH + r] - mx);
        float e1 = expf(sim[(t + 256) * RH + r] - mx);
        sim[t * RH + r] = e0; sim[(t + 256) * RH + r] = e1;
        red[t] = e0 + e1;
        __syncthreads();
        for (int s = 128; s > 0; s >>= 1) { if (t < s) red[t] += red[t + s]; __syncthreads(); }
        if (t == 0) ssum[r] = red[0];
        __syncthreads();
    }
    const float* Wfwd = ws + WS_WFWD;
    const float* Wbwd = ws + WS_WBWD;
#pragma unroll
    for (int h = 0; h < 2; ++h) {
        const int n = t + h * 256;
#pragma unroll
        for (int r = 0; r < RH; ++r) {
            float wl = sim[n * RH + r] / ssum[r];
            float v = rm[0 * RH + r] * Wbwd[(size_t)b * N_MEM * RH + n * RH + r]
                    + rm[1 * RH + r] * wl
                    + rm[2 * RH + r] * Wfwd[(size_t)b * N_MEM * RH + n * RH + r];
            sim[n * RH + r] = v;
            out[O_WRN + (size_t)b * N_MEM * RH + n * RH + r] = v;
        }
    }
    __syncthreads();
    // read_v[r][w] = sum_n M_new[n][w] * W_read_n[n][r]
    const int g = t >> 6, w = t & 63;
    float a0 = 0.f, a1 = 0.f, a2 = 0.f, a3 = 0.f;
    for (int n = g; n < N_MEM; n += 4) {
        float m = Mb[n * WD + w];
        a0 += m * sim[n * RH + 0]; a1 += m * sim[n * RH + 1];
        a2 += m * sim[n * RH + 2]; a3 += m * sim[n * RH + 3];
    }
    rv[g * (RH * WD) + 0 * WD + w] = a0;
    rv[g * (RH * WD) + 1 * WD + w] = a1;
    rv[g * (RH * WD) + 2 * WD + w] = a2;
    rv[g * (RH * WD) + 3 * WD + w] = a3;
    __syncthreads();
    float s = rv[t] + rv[256 + t] + rv[512 + t] + rv[768 + t];
    ws[WS_READV + (size_t)b * OUTD + t] = s;   // layout [b, r*64+w]
}

// ---------------- launcher ----------------
extern "C" void kernel_launch(void* const* d_in, const int* in_sizes, int n_in,
                              void* d_out, int out_size, void* d_ws, size_t ws_size,
                              hipStream_t stream)
{
    (void)in_sizes; (void)n_in; (void)out_size; (void)ws_size;
    const float* x_read  = (const float*)d_in[0];
    const float* x_write = (const float*)d_in[1];
    const float* M       = (const float*)d_in[2];
    const float* usage   = (const float*)d_in[3];
    const float* L       = (const float*)d_in[4];
    const float* Wp      = (const float*)d_in[5];
    const float* Wr      = (const float*)d_in[6];
    const float* Wwrt    = (const float*)d_in[7];
    const float* Wif     = (const float*)d_in[8];
    const float* Wro     = (const float*)d_in[9];
    float* out = (float*)d_out;
    float* ws  = (float*)d_ws;

    // interface GEMMs: [64,727] x [727,471] via fp32 WMMA
    dim3 g1((IFACE + 15) / 16, BS / 16);
    gemm_wmma_f32<<<g1, 32, 0, stream>>>(x_write, Wif, ws + WS_IFW, IFACE, CTRL, CTRL, IFACE, IFACE);
    gemm_wmma_f32<<<g1, 32, 0, stream>>>(x_read,  Wif, ws + WS_IFR, IFACE, CTRL, CTRL, IFACE, IFACE);

    prep_kernel<<<BS, 64, 0, stream>>>(ws);
    usage_kernel<<<(BS * N_MEM) / 256, 256, 0, stream>>>(usage, Wr, Wwrt, ws + WS_FG, out + O_U);
    alloc_kernel<<<BS, N_MEM, 0, stream>>>(out + O_U, ws + WS_WALLOC);
    write_kernel<<<BS, 256, 0, stream>>>(M, Wp, ws, out);
    zero_kernel<<<(BS * N_MEM * RH) / 256, 256, 0, stream>>>(ws + WS_WBWD, BS * N_MEM * RH);
    dim3 g8(N_MEM / 64, BS);
    l_kernel<<<g8, 256, 0, stream>>>(L, Wp, Wr, out + O_WW, out + O_L,
                                     ws + WS_WFWD, ws + WS_WBWD);
    read_kernel<<<BS, 256, 0, stream>>>(out + O_M, ws, out);

    // y = read_v[64,256] x W_read_out[256,256] via fp32 WMMA
    dim3 g10(OUTD / 16, BS / 16);
    gemm_wmma_f32<<<g10, 32, 0, stream>>>(ws + WS_READV, Wro, out + O_Y, OUTD, RH * WD, RH * WD, OUTD, OUTD);
}